// SAMBlock_73985106641413
// MI455X (gfx1250) — compile-verified
//
#include <hip/hip_runtime.h>
#include <math.h>

// CDNA5 / gfx1250: wave32, WMMA f32 16x16x4 (full-precision matrix pipe).
typedef __attribute__((ext_vector_type(2))) float v2f;
typedef __attribute__((ext_vector_type(8))) float v8f;

static constexpr int    kB   = 8;
static constexpr int    kC   = 128;
static constexpr int    kH   = 256;
static constexpr int    kW   = 256;
static constexpr int    kHW  = kH * kW;                 // 65536
static constexpr size_t kMap = (size_t)kB * kC * kHW;   // 67,108,864 elems (256 MiB fp32)

__device__ __forceinline__ v8f wmma_f32_k4(v8f acc, v2f a, v2f b) {
    // D = A(16x4) * B(4x16) + C, fp32 throughout.
    return __builtin_amdgcn_wmma_f32_16x16x4_f32(false, a, false, b,
                                                 (short)0, acc, false, false);
}

// ---------------------------------------------------------------------------
// Kernel 1: three fused 1x1 convs + PReLU.
//   map_q[b, o, s] = prelu( sum_c Wq[o,c] * x[b,c,s] + bq[o] ),  q in {f,g,h}
// Block = 64-spatial slab of one batch; x slab staged once in LDS transposed
// (xs[n][k]) so each B fragment is one conflict-free ds_load_b64 and x is read
// from HBM exactly once. 8 waves each own one 16-channel m-tile and compute a
// 16x64 strip with 4 accumulators (A fragment reused 4x per k-step).
// map_f / map_g are written SLICE-TRANSPOSED (fT[c][w][h]) so the attention
// kernel's A and B fragments become contiguous b64 loads; map_h stays [c][h][w].
// A 64-wide slab is always inside one H row, so the transpose index is cheap.
// ---------------------------------------------------------------------------
static constexpr int kTN   = 64;    // spatial columns per block
static constexpr int kXStr = 134;   // LDS k-stride (even -> 8B aligned; 134%64=6 -> bank-spread)

__global__ __launch_bounds__(256)
void conv3_prelu_kernel(const float* __restrict__ x,
                        const float* __restrict__ Wf, const float* __restrict__ bf,
                        const float* __restrict__ af,
                        const float* __restrict__ Wg, const float* __restrict__ bg,
                        const float* __restrict__ ag,
                        const float* __restrict__ Wh, const float* __restrict__ bh,
                        const float* __restrict__ ah,
                        float* __restrict__ mfT, float* __restrict__ mgT,
                        float* __restrict__ mh)
{
    __shared__ float xs[kTN * kXStr];       // 64 * 134 * 4 = 34,304 B

    const int tid  = threadIdx.x;
    const int lane = tid & 31;
    const int wave = tid >> 5;
    const int l15  = lane & 15;
    const int koff = (lane >> 4) << 1;      // lanes 16-31 carry K+2, K+3

    const int blk = blockIdx.x;             // kB * (kHW/kTN) = 8192 blocks
    const int b   = blk >> 10;              // 1024 slabs per batch
    const int s0  = (blk & 1023) * kTN;
    const float* xb = x + (size_t)b * kC * kHW + s0;

    // ---- stage X^T into LDS: xs[n*kXStr + k] = x[b, k, s0+n] -------------
    for (int i = tid; i < kC * (kTN / 4); i += 256) {   // 2048 float4 loads
        const int r  = i >> 4;              // k-row 0..127
        const int c4 = (i & 15) << 2;       // n-col 0,4,..,60
        const float4 v = *(const float4*)(xb + (size_t)r * kHW + c4);
        xs[(c4 + 0) * kXStr + r] = v.x;
        xs[(c4 + 1) * kXStr + r] = v.y;
        xs[(c4 + 2) * kXStr + r] = v.z;
        xs[(c4 + 3) * kXStr + r] = v.w;
    }
    __syncthreads();

    const int    m0    = wave << 4;         // 8 waves cover 128 out channels
    const int    mb    = m0 + ((lane >> 4) << 3);
    const float* bp    = &xs[(size_t)l15 * kXStr + koff];
    const size_t slice = (size_t)b * kC * kHW;
    const int    hrow  = s0 >> 8;           // constant: slab lies in one H row
    const int    wbase = (s0 & 255) + l15;

    auto gemm_q = [&](const float* __restrict__ Wm, const float* __restrict__ bias,
                      float alpha, float* __restrict__ om, bool transposeOut)
                      __attribute__((always_inline)) {
        const float* arow = Wm + (size_t)(m0 + l15) * kC + koff;
        v8f a0 = {}, a1 = {}, a2 = {}, a3 = {};
        for (int k = 0; k < kC; k += 4) {
            const v2f a  = *(const v2f*)(arow + k);                      // global b64
            const v2f q0 = *(const v2f*)(bp + (size_t)( 0) * kXStr + k); // ds b64
            const v2f q1 = *(const v2f*)(bp + (size_t)(16) * kXStr + k);
            const v2f q2 = *(const v2f*)(bp + (size_t)(32) * kXStr + k);
            const v2f q3 = *(const v2f*)(bp + (size_t)(48) * kXStr + k);
            a0 = wmma_f32_k4(a0, a, q0);
            a1 = wmma_f32_k4(a1, a, q1);
            a2 = wmma_f32_k4(a2, a, q2);
            a3 = wmma_f32_k4(a3, a, q3);
        }
        const v8f accs[4] = {a0, a1, a2, a3};
        if (transposeOut) {
            // fT/gT: slice index = c*kHW + w*kH + h
            #pragma unroll
            for (int nt = 0; nt < 4; ++nt) {
                #pragma unroll
                for (int j = 0; j < 8; ++j) {
                    const int c = mb + j;
                    float y = accs[nt][j] + bias[c];
                    y = (y >= 0.0f) ? y : alpha * y;
                    om[slice + (size_t)c * kHW + (size_t)(wbase + nt * 16) * kH + hrow] = y;
                }
            }
        } else {
            // h-map: slice index = c*kHW + s  (reference layout)
            #pragma unroll
            for (int nt = 0; nt < 4; ++nt) {
                #pragma unroll
                for (int j = 0; j < 8; ++j) {
                    const int c = mb + j;
                    float y = accs[nt][j] + bias[c];
                    y = (y >= 0.0f) ? y : alpha * y;
                    om[slice + (size_t)c * kHW + (size_t)(s0 + nt * 16 + l15)] = y;
                }
            }
        }
    };

    gemm_q(Wf, bf, *af, mfT, true);
    gemm_q(Wg, bg, *ag, mgT, true);
    gemm_q(Wh, bh, *ah, mh,  false);
}

// ---------------------------------------------------------------------------
// Kernel 2: per-(b,c) attention. One block per channel slice (1024 blocks).
//   S[w,v] = sum_h fT[w,h] * gT[v,h]      (stored transposed: ST[v][w])
//   softmax over v (= ST row index)       -> fully lane-coalesced passes
//   out[hr,v] = sum_w h[hr,w] * ST[v][w] + x[hr,v]
// 16x64 strips per wave; every fragment load is a contiguous b64.
// ---------------------------------------------------------------------------
__global__ __launch_bounds__(256)
void attention_kernel(const float* __restrict__ mfT, const float* __restrict__ mgT,
                      const float* __restrict__ mh, float* __restrict__ STbuf,
                      const float* __restrict__ x, float* __restrict__ out)
{
    const int tid  = threadIdx.x;
    const int lane = tid & 31;
    const int wave = tid >> 5;
    const int l15  = lane & 15;
    const int koff = (lane >> 4) << 1;

    const size_t base = (size_t)blockIdx.x * kHW;
    const float* fT = mfT + base;           // [w][h]
    const float* gT = mgT + base;           // [v][h]
    const float* hM = mh  + base;           // [h][w]
    float*       ST = STbuf + base;         // [v][w]

    // ---- Stage 1: ST[v, w] = sum_h fT[w,h] * gT[v,h] ---------------------
    // 64 strips = 16 w-tiles x 4 v-blocks(64 wide); 8 strips per wave.
    for (int st = wave; st < 64; st += 8) {
        const int w0 = (st >> 2) << 4;
        const int v0 = (st & 3) << 6;
        const float* arow = fT + (size_t)(w0 + l15) * kH + koff;  // A[m,k]=fT[w0+m,k]
        const float* brow = gT + (size_t)(v0 + l15) * kH + koff;  // B[k,n]=gT[v0+n,k]
        v8f a0 = {}, a1 = {}, a2 = {}, a3 = {};
        for (int k = 0; k < kH; k += 4) {
            const v2f a  = *(const v2f*)(arow + k);
            const v2f q0 = *(const v2f*)(brow + (size_t)( 0) * kH + k);
            const v2f q1 = *(const v2f*)(brow + (size_t)(16) * kH + k);
            const v2f q2 = *(const v2f*)(brow + (size_t)(32) * kH + k);
            const v2f q3 = *(const v2f*)(brow + (size_t)(48) * kH + k);
            a0 = wmma_f32_k4(a0, a, q0);
            a1 = wmma_f32_k4(a1, a, q1);
            a2 = wmma_f32_k4(a2, a, q2);
            a3 = wmma_f32_k4(a3, a, q3);
        }
        // D element (M = w0-tile row, N = v) -> ST[N][M]: per lane the 8
        // j-values are consecutive floats -> wide stores.
        const int mbW = w0 + ((lane >> 4) << 3);
        const v8f accs[4] = {a0, a1, a2, a3};
        #pragma unroll
        for (int nt = 0; nt < 4; ++nt) {
            float* sp = ST + (size_t)(v0 + nt * 16 + l15) * kW + mbW;
            #pragma unroll
            for (int j = 0; j < 8; ++j) sp[j] = accs[nt][j];
        }
    }
    __threadfence();
    __syncthreads();

    // ---- Stage 2: softmax over v for each w; thread t owns w = t ---------
    // ST[v*kW + t]: every pass reads/writes a contiguous 1KB line per v-step.
    {
        float m = -INFINITY;
        for (int v = 0; v < kW; ++v) m = fmaxf(m, ST[(size_t)v * kW + tid]);
        float s = 0.0f;
        for (int v = 0; v < kW; ++v) {
            const float e = __expf(ST[(size_t)v * kW + tid] - m);
            ST[(size_t)v * kW + tid] = e;
            s += e;
        }
        const float inv = 1.0f / s;
        for (int v = 0; v < kW; ++v) ST[(size_t)v * kW + tid] *= inv;
    }
    __threadfence();
    __syncthreads();

    // ---- Stage 3: out[hr, v] = sum_w h[hr, w] * ST[v, w] + x[hr, v] ------
    const float* xb = x + base;
    float*       ob = out + base;
    for (int st = wave; st < 64; st += 8) {
        const int m0 = (st >> 2) << 4;      // h-row tile
        const int v0 = (st & 3) << 6;       // v block
        const float* arow = hM + (size_t)(m0 + l15) * kW + koff;  // A[m,k]=h[m0+m,k]
        const float* brow = ST + (size_t)(v0 + l15) * kW + koff;  // B[k,n]=ST[v0+n,k]
        v8f a0 = {}, a1 = {}, a2 = {}, a3 = {};
        for (int k = 0; k < kW; k += 4) {
            const v2f a  = *(const v2f*)(arow + k);
            const v2f q0 = *(const v2f*)(brow + (size_t)( 0) * kW + k);
            const v2f q1 = *(const v2f*)(brow + (size_t)(16) * kW + k);
            const v2f q2 = *(const v2f*)(brow + (size_t)(32) * kW + k);
            const v2f q3 = *(const v2f*)(brow + (size_t)(48) * kW + k);
            a0 = wmma_f32_k4(a0, a, q0);
            a1 = wmma_f32_k4(a1, a, q1);
            a2 = wmma_f32_k4(a2, a, q2);
            a3 = wmma_f32_k4(a3, a, q3);
        }
        const int mb = m0 + ((lane >> 4) << 3);
        const v8f accs[4] = {a0, a1, a2, a3};
        #pragma unroll
        for (int nt = 0; nt < 4; ++nt) {
            #pragma unroll
            for (int j = 0; j < 8; ++j) {
                const size_t idx = (size_t)(mb + j) * kW + (size_t)(v0 + nt * 16 + l15);
                ob[idx] = accs[nt][j] + xb[idx];
            }
        }
    }
}

// ---------------------------------------------------------------------------
extern "C" void kernel_launch(void* const* d_in, const int* in_sizes, int n_in,
                              void* d_out, int out_size, void* d_ws, size_t ws_size,
                              hipStream_t stream)
{
    const float* x  = (const float*)d_in[0];
    const float* Wf = (const float*)d_in[1];
    const float* bf = (const float*)d_in[2];
    const float* af = (const float*)d_in[3];
    const float* Wg = (const float*)d_in[4];
    const float* bg = (const float*)d_in[5];
    const float* ag = (const float*)d_in[6];
    const float* Wh = (const float*)d_in[7];
    const float* bh = (const float*)d_in[8];
    const float* ah = (const float*)d_in[9];

    // Workspace layout (needs 4 * 256 MiB): fT | gT | h | ST
    float* ws  = (float*)d_ws;
    float* mfT = ws;
    float* mgT = mfT + kMap;
    float* mh  = mgT + kMap;
    float* STb = mh + kMap;

    conv3_prelu_kernel<<<dim3(kB * (kHW / kTN)), dim3(256), 0, stream>>>(
        x, Wf, bf, af, Wg, bg, ag, Wh, bh, ah, mfT, mgT, mh);

    attention_kernel<<<dim3(kB * kC), dim3(256), 0, stream>>>(
        mfT, mgT, mh, STb, x, (float*)d_out);
}